// SMPL_20461224198367
// MI455X (gfx1250) — compile-verified
//
#include <hip/hip_runtime.h>
#include <math.h>

typedef __attribute__((ext_vector_type(2))) float v2f;
typedef __attribute__((ext_vector_type(8))) float v8f;

#define JNT   24
#define NBETA 10
#define NPOSE 207
#define NV    6890
#define NCOL  (NV*3)      // 20670
#define LDA   224         // padded A row (K=218 -> 224, cols 218..223 zero)
#define NT4   323         // ceil(NCOL/64)

__device__ const int PAR[JNT] = {-1,0,0,0,1,2,3,4,5,6,7,8,9,9,9,12,13,14,16,17,18,19,20,21};

// ---------------- Kernel 1: Rodrigues + assemble A = [betas | lrotmin | 1 | 0-pad]
__global__ __launch_bounds__(256)
void k_prep(const float* __restrict__ pose, const float* __restrict__ betas,
            float* __restrict__ Amat, float* __restrict__ Rws, int Bsz) {
  int tid = blockIdx.x * blockDim.x + threadIdx.x;
  if (tid >= Bsz * JNT) return;
  int b = tid / JNT, j = tid - b * JNT;
  const float* pr = pose + (size_t)b * (JNT*3) + j * 3;
  float x = pr[0], y = pr[1], z = pr[2];
  float th = sqrtf(x*x + y*y + z*z) + 1e-8f;
  float inv = 1.0f / th;
  float rx = x*inv, ry = y*inv, rz = z*inv;
  float ct = cosf(th), st = sinf(th), oc = 1.0f - ct;
  float R[9];
  R[0]=ct+oc*rx*rx;    R[1]=oc*rx*ry-st*rz; R[2]=oc*rx*rz+st*ry;
  R[3]=oc*rx*ry+st*rz; R[4]=ct+oc*ry*ry;    R[5]=oc*ry*rz-st*rx;
  R[6]=oc*rx*rz-st*ry; R[7]=oc*ry*rz+st*rx; R[8]=ct+oc*rz*rz;
  float* rout = Rws + (size_t)(b*JNT + j) * 9;
#pragma unroll
  for (int e = 0; e < 9; ++e) rout[e] = R[e];
  if (j >= 1) {
    float* arow = Amat + (size_t)b * LDA + 10 + (j-1) * 9;
#pragma unroll
    for (int e = 0; e < 9; ++e)
      arow[e] = R[e] - ((e==0 || e==4 || e==8) ? 1.0f : 0.0f);
  } else {
    float* arow = Amat + (size_t)b * LDA;
#pragma unroll
    for (int k = 0; k < NBETA; ++k) arow[k] = betas[(size_t)b*NBETA + k];
    arow[217] = 1.0f;
#pragma unroll
    for (int k = 218; k < LDA; ++k) arow[k] = 0.0f;
  }
}

// ---------------- Kernel 2: JS[j,c,k] = sum_v Jreg[j,v]*shapedirs[v,c,k]; JT = Jreg @ v_template
__global__ __launch_bounds__(256)
void k_js(const float* __restrict__ Jreg, const float* __restrict__ shapedirs,
          const float* __restrict__ v_template, float* __restrict__ JS, float* __restrict__ JT) {
  int blk = blockIdx.x;          // 0..71 = j*3+c
  int j = blk / 3, c = blk - j * 3;
  int t = threadIdx.x;
  float acc[11];
#pragma unroll
  for (int q = 0; q < 11; ++q) acc[q] = 0.0f;
  const float* jrow = Jreg + (size_t)j * NV;
  for (int v = t; v < NV; v += 256) {
    float r = jrow[v];
    const float* sd = shapedirs + (size_t)(v*3 + c) * NBETA;
#pragma unroll
    for (int k = 0; k < NBETA; ++k) acc[k] += r * sd[k];
    acc[10] += r * v_template[v*3 + c];
  }
  __shared__ float sh[256];
  for (int q = 0; q < 11; ++q) {
    sh[t] = acc[q];
    __syncthreads();
    for (int s = 128; s > 0; s >>= 1) {
      if (t < s) sh[t] += sh[t + s];
      __syncthreads();
    }
    if (t == 0) {
      if (q < NBETA) JS[(size_t)blk * NBETA + q] = sh[0];
      else           JT[blk] = sh[0];
    }
    __syncthreads();
  }
}

// ---------------- Kernel 3: Jpos[b,j,c] = JT[j,c] + betas[b]·JS[j,c,:]
__global__ __launch_bounds__(256)
void k_jpos(const float* __restrict__ betas, const float* __restrict__ JS,
            const float* __restrict__ JT, float* __restrict__ Jpos, int Bsz) {
  int tid = blockIdx.x * blockDim.x + threadIdx.x;
  if (tid >= Bsz * JNT) return;
  int b = tid / JNT, j = tid - b * JNT;
  const float* bt = betas + (size_t)b * NBETA;
#pragma unroll
  for (int c = 0; c < 3; ++c) {
    float s = JT[j*3 + c];
    const float* js = JS + (size_t)(j*3 + c) * NBETA;
#pragma unroll
    for (int k = 0; k < NBETA; ++k) s += bt[k] * js[k];
    Jpos[(size_t)(b*JNT + j)*3 + c] = s;
  }
}

// ---------------- Kernel 4: kinematic chain -> world transforms, corrected G (3x4 per joint)
__global__ __launch_bounds__(64)
void k_chain(const float* __restrict__ Rws, const float* __restrict__ Jws,
             float* __restrict__ Gw, float* __restrict__ G, int Bsz) {
  int b = blockIdx.x * blockDim.x + threadIdx.x;
  if (b >= Bsz) return;
  const float* Rb = Rws + (size_t)b * (JNT*9);
  const float* Jb = Jws + (size_t)b * (JNT*3);
  float* Wb = Gw + (size_t)b * (JNT*12);
  float* Gb = G  + (size_t)b * (JNT*12);
  for (int i = 0; i < JNT; ++i) {
    const float* Ri = Rb + i * 9;
    float jx = Jb[i*3+0], jy = Jb[i*3+1], jz = Jb[i*3+2];
    float tx, ty, tz;
    int p = PAR[i];
    if (i == 0) { tx = jx; ty = jy; tz = jz; }
    else { tx = jx - Jb[p*3+0]; ty = jy - Jb[p*3+1]; tz = jz - Jb[p*3+2]; }
    float W[12];
    if (i == 0) {
#pragma unroll
      for (int r = 0; r < 3; ++r) {
        W[r*4+0] = Ri[r*3+0]; W[r*4+1] = Ri[r*3+1]; W[r*4+2] = Ri[r*3+2];
      }
      W[3] = tx; W[7] = ty; W[11] = tz;
    } else {
      const float* Pm = Wb + p * 12;
#pragma unroll
      for (int r = 0; r < 3; ++r) {
        float p0 = Pm[r*4+0], p1 = Pm[r*4+1], p2 = Pm[r*4+2], p3 = Pm[r*4+3];
        W[r*4+0] = p0*Ri[0] + p1*Ri[3] + p2*Ri[6];
        W[r*4+1] = p0*Ri[1] + p1*Ri[4] + p2*Ri[7];
        W[r*4+2] = p0*Ri[2] + p1*Ri[5] + p2*Ri[8];
        W[r*4+3] = p0*tx + p1*ty + p2*tz + p3;
      }
    }
    float* wout = Wb + i * 12;
#pragma unroll
    for (int e = 0; e < 12; ++e) wout[e] = W[e];
    float* gout = Gb + i * 12;
#pragma unroll
    for (int r = 0; r < 3; ++r) {
      float corr = W[r*4+0]*jx + W[r*4+1]*jy + W[r*4+2]*jz;
      gout[r*4+0] = W[r*4+0]; gout[r*4+1] = W[r*4+1]; gout[r*4+2] = W[r*4+2];
      gout[r*4+3] = W[r*4+3] - corr;
    }
  }
}

// ---------------- Kernel 5: v_posed = A(Bsz x 220) @ Bmat(220 x 20670), V_WMMA_F32_16X16X4_F32
// K layout: rows 0..9 = shapedirs, 10..216 = posedirs, 217 = v_template, 218..219 = zero (A padded).
// Phase 1: k0=0,4,8  (branchless pointer select between shapedirs/posedirs)
// Phase 2: k0=12..212 (pure posedirs, software-pipelined)
// Phase 3: k0=216     (posedirs p=206 + template; hi-lane rows 218/219 killed by A zero-pad)
__global__ __launch_bounds__(256)
void k_blend_gemm(const float* __restrict__ Amat, const float* __restrict__ sd,
                  const float* __restrict__ pd, const float* __restrict__ vt,
                  float* __restrict__ mesh, int Bsz) {
  const int lane = threadIdx.x & 31;
  const int wid  = blockIdx.x * (blockDim.x >> 5) + (threadIdx.x >> 5);
  const int mtiles = (Bsz + 15) >> 4;
  if (wid >= mtiles * NT4) return;
  const int mt = wid / NT4;
  const int nt = wid - mt * NT4;
  const int m0 = mt << 4;
  const int nb = nt << 6;
  const int lo16 = lane & 15;
  const int hi   = lane >> 4;
  const int hi2  = hi << 1;
  int m = m0 + lo16;
  int mload = m < Bsz ? m : (Bsz - 1);
  const int n0 = nb + lo16, n1 = n0 + 16, n2 = n0 + 32, n3 = n0 + 48;
  // clamped column indices: OOB lanes load valid memory; their stores are masked.
  const int c0 = n0 < NCOL ? n0 : (NCOL - 1);
  const int c1 = n1 < NCOL ? n1 : (NCOL - 1);
  const int c2 = n2 < NCOL ? n2 : (NCOL - 1);
  const int c3 = n3 < NCOL ? n3 : (NCOL - 1);

  v8f acc0 = {}, acc1 = {}, acc2 = {}, acc3 = {};
  const float* arow = Amat + (size_t)mload * LDA;

  // -------- Phase 1: k0 = 0,4,8
#pragma unroll
  for (int k0 = 0; k0 <= 8; k0 += 4) {
    const int kk = k0 + hi2;
    v2f a = *(const v2f*)(arow + kk);
    const float* p0 = (kk < 10) ? (sd + (size_t)c0 * NBETA + kk) : (pd + (size_t)c0 * NPOSE + (kk - 10));
    const float* p1 = (kk < 10) ? (sd + (size_t)c1 * NBETA + kk) : (pd + (size_t)c1 * NPOSE + (kk - 10));
    const float* p2 = (kk < 10) ? (sd + (size_t)c2 * NBETA + kk) : (pd + (size_t)c2 * NPOSE + (kk - 10));
    const float* p3 = (kk < 10) ? (sd + (size_t)c3 * NBETA + kk) : (pd + (size_t)c3 * NPOSE + (kk - 10));
    v2f b0; b0.x = p0[0]; b0.y = p0[1];
    v2f b1; b1.x = p1[0]; b1.y = p1[1];
    v2f b2; b2.x = p2[0]; b2.y = p2[1];
    v2f b3; b3.x = p3[0]; b3.y = p3[1];
    acc0 = __builtin_amdgcn_wmma_f32_16x16x4_f32(false, a, false, b0, (short)0, acc0, false, false);
    acc1 = __builtin_amdgcn_wmma_f32_16x16x4_f32(false, a, false, b1, (short)0, acc1, false, false);
    acc2 = __builtin_amdgcn_wmma_f32_16x16x4_f32(false, a, false, b2, (short)0, acc2, false, false);
    acc3 = __builtin_amdgcn_wmma_f32_16x16x4_f32(false, a, false, b3, (short)0, acc3, false, false);
  }

  // -------- Phase 2: k0 = 12..212 (51 iterations), double-buffered
  {
    const float* pa  = arow + 12 + hi2;                       // 8B aligned (even offset)
    const float* pb0 = pd + (size_t)c0 * NPOSE + 2 + hi2;     // p = kk-10, kk = 12+hi2
    const float* pb1 = pd + (size_t)c1 * NPOSE + 2 + hi2;
    const float* pb2 = pd + (size_t)c2 * NPOSE + 2 + hi2;
    const float* pb3 = pd + (size_t)c3 * NPOSE + 2 + hi2;

    v2f a_c = *(const v2f*)pa;
    v2f b0_c; b0_c.x = pb0[0]; b0_c.y = pb0[1];
    v2f b1_c; b1_c.x = pb1[0]; b1_c.y = pb1[1];
    v2f b2_c; b2_c.x = pb2[0]; b2_c.y = pb2[1];
    v2f b3_c; b3_c.x = pb3[0]; b3_c.y = pb3[1];

    for (int it = 0; it < 50; ++it) {
      pa += 4; pb0 += 4; pb1 += 4; pb2 += 4; pb3 += 4;
      v2f a_n = *(const v2f*)pa;
      v2f b0_n; b0_n.x = pb0[0]; b0_n.y = pb0[1];
      v2f b1_n; b1_n.x = pb1[0]; b1_n.y = pb1[1];
      v2f b2_n; b2_n.x = pb2[0]; b2_n.y = pb2[1];
      v2f b3_n; b3_n.x = pb3[0]; b3_n.y = pb3[1];
      acc0 = __builtin_amdgcn_wmma_f32_16x16x4_f32(false, a_c, false, b0_c, (short)0, acc0, false, false);
      acc1 = __builtin_amdgcn_wmma_f32_16x16x4_f32(false, a_c, false, b1_c, (short)0, acc1, false, false);
      acc2 = __builtin_amdgcn_wmma_f32_16x16x4_f32(false, a_c, false, b2_c, (short)0, acc2, false, false);
      acc3 = __builtin_amdgcn_wmma_f32_16x16x4_f32(false, a_c, false, b3_c, (short)0, acc3, false, false);
      a_c = a_n; b0_c = b0_n; b1_c = b1_n; b2_c = b2_n; b3_c = b3_n;
    }
    acc0 = __builtin_amdgcn_wmma_f32_16x16x4_f32(false, a_c, false, b0_c, (short)0, acc0, false, false);
    acc1 = __builtin_amdgcn_wmma_f32_16x16x4_f32(false, a_c, false, b1_c, (short)0, acc1, false, false);
    acc2 = __builtin_amdgcn_wmma_f32_16x16x4_f32(false, a_c, false, b2_c, (short)0, acc2, false, false);
    acc3 = __builtin_amdgcn_wmma_f32_16x16x4_f32(false, a_c, false, b3_c, (short)0, acc3, false, false);
  }

  // -------- Phase 3: k0 = 216 (rows 216=posedirs[206], 217=template; 218/219 zeroed by A pad)
  {
    const int kk = 216 + hi2;
    v2f a = *(const v2f*)(arow + kk);
    v2f b0; b0.x = pd[(size_t)c0 * NPOSE + 206]; b0.y = vt[c0];
    v2f b1; b1.x = pd[(size_t)c1 * NPOSE + 206]; b1.y = vt[c1];
    v2f b2; b2.x = pd[(size_t)c2 * NPOSE + 206]; b2.y = vt[c2];
    v2f b3; b3.x = pd[(size_t)c3 * NPOSE + 206]; b3.y = vt[c3];
    acc0 = __builtin_amdgcn_wmma_f32_16x16x4_f32(false, a, false, b0, (short)0, acc0, false, false);
    acc1 = __builtin_amdgcn_wmma_f32_16x16x4_f32(false, a, false, b1, (short)0, acc1, false, false);
    acc2 = __builtin_amdgcn_wmma_f32_16x16x4_f32(false, a, false, b2, (short)0, acc2, false, false);
    acc3 = __builtin_amdgcn_wmma_f32_16x16x4_f32(false, a, false, b3, (short)0, acc3, false, false);
  }

#pragma unroll
  for (int r = 0; r < 8; ++r) {
    int row = m0 + (hi << 3) + r;
    if (row < Bsz) {
      float* base = mesh + (size_t)row * NCOL;
      if (n0 < NCOL) base[n0] = acc0[r];
      if (n1 < NCOL) base[n1] = acc1[r];
      if (n2 < NCOL) base[n2] = acc2[r];
      if (n3 < NCOL) base[n3] = acc3[r];
    }
  }
}

// ---------------- Kernel 6: LBS in-place on mesh + translation (float4 G/weight reads)
__global__ __launch_bounds__(256)
void k_lbs(float* __restrict__ mesh, const float* __restrict__ G,
           const float* __restrict__ weights, const float* __restrict__ trans, int Bsz) {
  long long tid = (long long)blockIdx.x * blockDim.x + threadIdx.x;
  long long total = (long long)Bsz * NV;
  if (tid >= total) return;
  int b = (int)(tid / NV);
  int v = (int)(tid - (long long)b * NV);
  float* pv = mesh + (size_t)b * NCOL + (size_t)v * 3;
  float px = pv[0], py = pv[1], pz = pv[2];

  float wbuf[JNT];
  {
    const float4* w4 = (const float4*)(weights + (size_t)v * JNT);  // 16B aligned: v*96B
#pragma unroll
    for (int q = 0; q < 6; ++q) {
      float4 w = w4[q];
      wbuf[q*4+0] = w.x; wbuf[q*4+1] = w.y; wbuf[q*4+2] = w.z; wbuf[q*4+3] = w.w;
    }
  }
  float T[12];
#pragma unroll
  for (int e = 0; e < 12; ++e) T[e] = 0.0f;
  const float4* G4 = (const float4*)(G + (size_t)b * (JNT*12));     // 16B aligned: j*48B
#pragma unroll 4
  for (int j = 0; j < JNT; ++j) {
    float w = wbuf[j];
    float4 ga = G4[j*3+0], gb = G4[j*3+1], gc = G4[j*3+2];
    T[0] += w*ga.x; T[1] += w*ga.y; T[2]  += w*ga.z; T[3]  += w*ga.w;
    T[4] += w*gb.x; T[5] += w*gb.y; T[6]  += w*gb.z; T[7]  += w*gb.w;
    T[8] += w*gc.x; T[9] += w*gc.y; T[10] += w*gc.z; T[11] += w*gc.w;
  }
  float ox = T[0]*px + T[1]*py + T[2] *pz + T[3]  + trans[(size_t)b*3 + 0];
  float oy = T[4]*px + T[5]*py + T[6] *pz + T[7]  + trans[(size_t)b*3 + 1];
  float oz = T[8]*px + T[9]*py + T[10]*pz + T[11] + trans[(size_t)b*3 + 2];
  pv[0] = ox; pv[1] = oy; pv[2] = oz;
}

// ---------------- Kernel 7: joints = einsum('bvc,jv->bjc', mesh, joint_regressor)
__global__ __launch_bounds__(256)
void k_joints(const float* __restrict__ mesh, const float* __restrict__ jr,
              float* __restrict__ jout, int Bsz) {
  int blk = blockIdx.x;
  int b = blk / JNT, j = blk - b * JNT;
  int t = threadIdx.x;
  float a0 = 0.0f, a1 = 0.0f, a2 = 0.0f;
  const float* jrow = jr + (size_t)j * NV;
  const float* mb = mesh + (size_t)b * NCOL;
  for (int v = t; v < NV; v += 256) {
    float r = jrow[v];
    const float* p = mb + (size_t)v * 3;
    a0 += r * p[0]; a1 += r * p[1]; a2 += r * p[2];
  }
  __shared__ float sh[256];
  float acc[3] = {a0, a1, a2};
  for (int q = 0; q < 3; ++q) {
    sh[t] = acc[q];
    __syncthreads();
    for (int s = 128; s > 0; s >>= 1) {
      if (t < s) sh[t] += sh[t + s];
      __syncthreads();
    }
    if (t == 0) jout[(size_t)(b*JNT + j)*3 + q] = sh[0];
    __syncthreads();
  }
}

extern "C" void kernel_launch(void* const* d_in, const int* in_sizes, int n_in,
                              void* d_out, int out_size, void* d_ws, size_t ws_size,
                              hipStream_t stream) {
  const float* betas      = (const float*)d_in[0];
  const float* pose       = (const float*)d_in[1];
  const float* trans      = (const float*)d_in[2];
  const float* Jreg       = (const float*)d_in[3];
  const float* jreg2      = (const float*)d_in[4];
  const float* weights    = (const float*)d_in[5];
  const float* posedirs   = (const float*)d_in[6];
  const float* v_template = (const float*)d_in[7];
  const float* shapedirs  = (const float*)d_in[8];
  const int Bsz = in_sizes[0] / NBETA;

  float* ws = (float*)d_ws;
  float* Amat = ws;                                   // Bsz * 224
  float* Rws  = Amat + (size_t)Bsz * LDA;             // Bsz * 216
  float* JS   = Rws  + (size_t)Bsz * (JNT*9);         // 720
  float* JT   = JS   + 72 * NBETA;                    // 72
  float* Jpos = JT   + 72;                            // Bsz * 72
  float* Gw   = Jpos + (size_t)Bsz * (JNT*3);         // Bsz * 288
  float* G    = Gw   + (size_t)Bsz * (JNT*12);        // Bsz * 288

  float* mesh = (float*)d_out;                        // Bsz * 20670
  float* jout = mesh + (size_t)Bsz * NCOL;            // Bsz * 72

  {
    int n = Bsz * JNT;
    k_prep<<<(n + 255) / 256, 256, 0, stream>>>(pose, betas, Amat, Rws, Bsz);
  }
  k_js<<<72, 256, 0, stream>>>(Jreg, shapedirs, v_template, JS, JT);
  {
    int n = Bsz * JNT;
    k_jpos<<<(n + 255) / 256, 256, 0, stream>>>(betas, JS, JT, Jpos, Bsz);
  }
  k_chain<<<(Bsz + 63) / 64, 64, 0, stream>>>(Rws, Jpos, Gw, G, Bsz);
  {
    int mtiles = (Bsz + 15) / 16;
    int waves = mtiles * NT4;
    k_blend_gemm<<<(waves + 7) / 8, 256, 0, stream>>>(Amat, shapedirs, posedirs,
                                                      v_template, mesh, Bsz);
  }
  {
    long long n = (long long)Bsz * NV;
    int blocks = (int)((n + 255) / 256);
    k_lbs<<<blocks, 256, 0, stream>>>(mesh, G, weights, trans, Bsz);
  }
  k_joints<<<Bsz * JNT, 256, 0, stream>>>(mesh, jreg2, jout, Bsz);
}